// NCC_56109452755311
// MI455X (gfx1250) — compile-verified
//
#include <hip/hip_runtime.h>

typedef __attribute__((ext_vector_type(2))) float v2f;
typedef __attribute__((ext_vector_type(8))) float v8f;

#define N 160
#define TILE 16
#define HALO 4
#define TIN 24            // TILE + 2*HALO
#define NT 10             // tiles per spatial dim
#define NCHUNK 4
#define CHD 40            // depth outputs per chunk
#define NWG 800           // 2 batches * 4 chunks * 10 * 10
#define WSZ 729.0f
#define INV_WSZ (1.0f / 729.0f)

// shared-memory layout (float offsets)
#define OFF_I    0                  // 24*24 = 576
#define OFF_J    576                // 576
#define OFF_SW   1152               // 5 * 16 * 24 = 1920
#define OFF_RING 3072               // 9 slots * 5 fields * 256 = 11520
#define SMEM_FLOATS 14592           // 58368 bytes

__global__ __launch_bounds__(256) void ncc_main_kernel(const float* __restrict__ I,
                                                       const float* __restrict__ J,
                                                       float* __restrict__ partial) {
  __shared__ float sm[SMEM_FLOATS];
  const int t = threadIdx.x;
  const int wg = blockIdx.x;

  const int b = wg / (NCHUNK * NT * NT);
  int rem = wg % (NCHUNK * NT * NT);
  const int chunk = rem / (NT * NT);
  rem = rem % (NT * NT);
  const int ty = rem / NT, tx = rem % NT;
  const int x0 = tx * TILE - HALO;   // leftmost input x of tile (may be <0)
  const int y0 = ty * TILE - HALO;
  const int zo0 = chunk * CHD;       // output z range [zo0, zo1)
  const int zo1 = zo0 + CHD;

  const int lane = t & 31;
  const int waveId = t >> 5;
  const int lo = lane & 15;          // WMMA: column (N) / row-lane
  const int hi = lane >> 4;          // WMMA: lane-half selector

  // Constant banded B matrix (K=4 chunks of the 24x16 ones-band):
  // B[h][n] = 1 iff n <= h <= n+8.  Layout for V_WMMA_F32_16X16X4_F32 B (4x16):
  // vgpr0: K = k0 + 2*hi, vgpr1: K = k0 + 2*hi + 1, column = lane&15.
  v2f bmat[6];
#pragma unroll
  for (int c = 0; c < 6; ++c) {
    const int h0 = 4 * c + 2 * hi;
    bmat[c].x = (h0     >= lo && h0     <= lo + 8) ? 1.0f : 0.0f;
    bmat[c].y = (h0 + 1 >= lo && h0 + 1 <= lo + 8) ? 1.0f : 0.0f;
  }

  // zero the z-ring
  for (int e = t; e < 9 * 5 * 256; e += 256) sm[OFF_RING + e] = 0.0f;
  __syncthreads();

  float running[5] = {0.f, 0.f, 0.f, 0.f, 0.f};
  float ccsum = 0.0f;

  for (int z = zo0 - 4; z < zo1 + 4; ++z) {
    const int slot = (z + 36) % 9;     // z >= -4, keep mod positive
    const int ringBase = OFF_RING + slot * 5 * 256;

    // ---- step 1: snapshot ring-old (slice z-9) + load zero-padded I/J tiles
    float oldv[5];
#pragma unroll
    for (int f = 0; f < 5; ++f) oldv[f] = sm[ringBase + f * 256 + t];

    const bool zok = (z >= 0) && (z < N);
    for (int e = t; e < TIN * TIN; e += 256) {
      const int hl = e / TIN, wl = e - hl * TIN;
      const int gx = x0 + wl, gy = y0 + hl;
      float vi = 0.0f, vj = 0.0f;
      if (zok && gx >= 0 && gx < N && gy >= 0 && gy < N) {
        const int idx = (((b * N) + z) * N + gy) * N + gx;   // < 8.2M, fits int
        vi = I[idx];
        vj = J[idx];
      }
      sm[OFF_I + e] = vi;
      sm[OFF_J + e] = vj;
    }

    // Prefetch next z-slice (one prefetch per tile row -> global_prefetch_b8),
    // hides the next iteration's global->LDS latency behind this slice's math.
    if (t < TIN) {
      const int zn = z + 1;
      const int gy = y0 + t;
      if (zn >= 0 && zn < N && gy >= 0 && gy < N) {
        const int gx0 = (x0 < 0) ? 0 : x0;
        const int idx = (((b * N) + zn) * N + gy) * N + gx0;
        __builtin_prefetch(&I[idx], 0, 3);
        __builtin_prefetch(&J[idx], 0, 3);
      }
    }
    __syncthreads();

    // ---- step 2: W-direction sliding box sum of the 5 derived fields
    // produces Sw[f][w 0..15][h 0..23] in LDS (h contiguous for v2f A loads)
    for (int p = t; p < 16 * TIN; p += 256) {
      const int w = p & 15;
      const int hrow = p >> 4;
      float sI = 0.f, sJ = 0.f, sI2 = 0.f, sJ2 = 0.f, sIJ = 0.f;
#pragma unroll
      for (int k = 0; k < 9; ++k) {
        const float x = sm[OFF_I + hrow * TIN + w + k];
        const float y = sm[OFF_J + hrow * TIN + w + k];
        sI += x; sJ += y;
        sI2 = fmaf(x, x, sI2);
        sJ2 = fmaf(y, y, sJ2);
        sIJ = fmaf(x, y, sIJ);
      }
      const int base = OFF_SW + w * TIN + hrow;
      sm[base + 0 * 16 * TIN] = sI;
      sm[base + 1 * 16 * TIN] = sJ;
      sm[base + 2 * 16 * TIN] = sI2;
      sm[base + 3 * 16 * TIN] = sJ2;
      sm[base + 4 * 16 * TIN] = sIJ;
    }
    __syncthreads();

    // ---- step 3: H-direction box sum as banded GEMM on the matrix pipe.
    // Out(16x16)[w][ho] = sum_h Sw[w][h] * Band[h][ho], K=24 in six K=4 steps.
    // Waves 0..4 each own one field; EXEC all-ones within each wave.
    if (waveId < 5) {
      const int f = waveId;
      v8f acc = {};
#pragma unroll
      for (int c = 0; c < 6; ++c) {
        const int k0 = 4 * c;
        // A (16x4): vgpr0 = Sw[w=lo][k0+2*hi], vgpr1 = Sw[lo][k0+2*hi+1]
        const int sbase = OFF_SW + (f * 16 + lo) * TIN + (k0 + 2 * hi);
        v2f a;
        a.x = sm[sbase];
        a.y = sm[sbase + 1];
#if __has_builtin(__builtin_amdgcn_wmma_f32_16x16x4_f32)
        acc = __builtin_amdgcn_wmma_f32_16x16x4_f32(false, a, false, bmat[c],
                                                    (short)0, acc, false, false);
#else
        // scalar fallback (keeps compile alive if builtin name probes wrong)
#pragma unroll
        for (int r = 0; r < 8; ++r) {
          const int m = r + 8 * hi;
          acc[r] += a.x * ((lo <= (k0 + 0) && (k0 + 0) <= lo + 8) ? 1.f : 0.f) * sm[OFF_SW + (f * 16 + m) * TIN + k0];
        }
#endif
      }
      // scatter D (16x16 f32 layout: vgpr r -> row r + 8*hi, col = lane&15)
#pragma unroll
      for (int r = 0; r < 8; ++r) {
        const int m = r + 8 * hi;
        sm[ringBase + f * 256 + m * 16 + lo] = acc[r];
      }
    }
    __syncthreads();

    // ---- step 4: z-direction sliding update + emit cc for zo = z - 4
#pragma unroll
    for (int f = 0; f < 5; ++f) {
      const float nv = sm[ringBase + f * 256 + t];
      running[f] += nv - oldv[f];
    }
    const int zo = z - 4;
    if (zo >= zo0 && zo < zo1) {
      const float S0 = running[0], S1 = running[1];
      const float S2 = running[2], S3 = running[3], S4 = running[4];
      // Constant-reciprocal instead of IEEE divide (avoids v_div_scale/fmas chain)
      const float uI = S0 * INV_WSZ, uJ = S1 * INV_WSZ;
      const float cross = S4 - uJ * S0 - uI * S1 + uI * uJ * WSZ;
      const float Ivar = S2 - 2.0f * uI * S0 + uI * uI * WSZ;
      const float Jvar = S3 - 2.0f * uJ * S1 + uJ * uJ * WSZ;
      const float denom = Ivar * Jvar + 1e-5f;
#if __has_builtin(__builtin_amdgcn_rcpf)
      ccsum += cross * cross * __builtin_amdgcn_rcpf(denom);  // v_rcp_f32, ~1 ulp
#else
      ccsum += cross * cross / denom;
#endif
    }
  }

  // ---- deterministic workgroup reduction (reuse tile LDS)
  __syncthreads();
  sm[t] = ccsum;
  __syncthreads();
  for (int s = 128; s > 0; s >>= 1) {
    if (t < s) sm[t] += sm[t + s];
    __syncthreads();
  }
  if (t == 0) partial[wg] = sm[0];
}

__global__ __launch_bounds__(256) void ncc_reduce_kernel(const float* __restrict__ partial,
                                                         float* __restrict__ out) {
  __shared__ float red[256];
  const int t = threadIdx.x;
  float s = 0.0f;
  for (int i = t; i < NWG; i += 256) s += partial[i];
  red[t] = s;
  __syncthreads();
  for (int k = 128; k > 0; k >>= 1) {
    if (t < k) red[t] += red[t + k];
    __syncthreads();
  }
  if (t == 0) out[0] = -red[0] / 8192000.0f;   // 2 * 160^3 voxels
}

extern "C" void kernel_launch(void* const* d_in, const int* in_sizes, int n_in,
                              void* d_out, int out_size, void* d_ws, size_t ws_size,
                              hipStream_t stream) {
  (void)in_sizes; (void)n_in; (void)out_size; (void)ws_size;
  const float* I = (const float*)d_in[0];   // y_true
  const float* J = (const float*)d_in[1];   // y_pred
  float* partial = (float*)d_ws;            // NWG floats of scratch
  float* out = (float*)d_out;

  ncc_main_kernel<<<NWG, 256, 0, stream>>>(I, J, partial);
  ncc_reduce_kernel<<<1, 256, 0, stream>>>(partial, out);
}